// WindowAttention_61933428414921
// MI455X (gfx1250) — compile-verified
//
#include <hip/hip_runtime.h>
#include <math.h>

typedef __attribute__((ext_vector_type(2))) float v2f;
typedef __attribute__((ext_vector_type(8))) float v8f;

#define KDIM   512
#define KNH    16
#define KHEAD  32
#define KH     64
#define KW     64
#define KG     32     // windows per image
#define KN     128    // tokens per window
#define K2N    256    // keys per window (2 rolls)
#define KSHIFT 4

static __device__ __forceinline__ v8f wmma_f32(v2f a, v2f b, v8f c) {
  // V_WMMA_F32_16X16X4_F32: D = A(16x4) * B(4x16) + C(16x16)
  return __builtin_amdgcn_wmma_f32_16x16x4_f32(false, a, false, b, (short)0, c,
                                               false, false);
}

// ---------------------------------------------------------------------------
// Generic GEMM: C[M,N] = A[M,512] * W[512,N] + bias[N]
// 256 threads (8 waves). Block tile 128(M) x 64(N), K-chunks of 64.
// B tile stored TRANSPOSED in LDS so each B-fragment is one contiguous b64.
// Register double-buffer: global loads of chunk i+1 overlap compute of i.
// ---------------------------------------------------------------------------
__global__ __launch_bounds__(256) void gemm_bias_kernel(
    const float* __restrict__ A, const float* __restrict__ Wm,
    const float* __restrict__ bias, float* __restrict__ C, int M, int N) {
  __shared__ float As[128][68];    // padded: read banks 4*l16, 16B aligned
  __shared__ float BsT[64][66];    // [n][k], padded: read banks 2*l16, 8B ok

  const int nBlocks = N >> 6;
  const int mBlock  = blockIdx.x / nBlocks;
  const int nBlock  = blockIdx.x % nBlocks;
  const int rowBase = mBlock << 7;
  const int colBase = nBlock << 6;

  const int tid  = threadIdx.x;
  const int wave = tid >> 5;
  const int lane = tid & 31;
  const int l16  = lane & 15;
  const int half = lane >> 4;

  v8f acc[4];
  acc[0] = {}; acc[1] = {}; acc[2] = {}; acc[3] = {};

  const int arow = tid >> 1;            // 0..127
  const int acol = (tid & 1) << 5;      // 0 or 32
  const int brow = tid >> 2;            // 0..63  (k within chunk)
  const int bcol = (tid & 3) << 4;      // 0,16,32,48 (n within tile)

  const float* ag0 = A + (size_t)(rowBase + arow) * KDIM + acol;
  const float* bg0 = Wm + (size_t)brow * N + colBase + bcol;

  float4 ra[8], rb[4];
#pragma unroll
  for (int i = 0; i < 8; ++i) ra[i] = ((const float4*)ag0)[i];
#pragma unroll
  for (int i = 0; i < 4; ++i) rb[i] = ((const float4*)bg0)[i];

  for (int chunk = 0; chunk < 8; ++chunk) {
    // regs -> LDS
#pragma unroll
    for (int i = 0; i < 8; ++i)
      *(float4*)&As[arow][acol + 4 * i] = ra[i];
#pragma unroll
    for (int i = 0; i < 4; ++i) {
      BsT[bcol + 4 * i + 0][brow] = rb[i].x;
      BsT[bcol + 4 * i + 1][brow] = rb[i].y;
      BsT[bcol + 4 * i + 2][brow] = rb[i].z;
      BsT[bcol + 4 * i + 3][brow] = rb[i].w;
    }
    __syncthreads();

    if (chunk < 7) {  // prefetch next K-chunk into registers (overlaps WMMA)
      const float* ag = ag0 + (chunk + 1) * 64;
      const float* bg = bg0 + (size_t)(chunk + 1) * 64 * N;
#pragma unroll
      for (int i = 0; i < 8; ++i) ra[i] = ((const float4*)ag)[i];
#pragma unroll
      for (int i = 0; i < 4; ++i) rb[i] = ((const float4*)bg)[i];
    }

#pragma unroll
    for (int kk = 0; kk < 16; ++kk) {
      // A frag: lane -> M=l16, K = 4kk + 2*half + {0,1}
      v2f a = *(const v2f*)&As[wave * 16 + l16][4 * kk + 2 * half];
#pragma unroll
      for (int nt = 0; nt < 4; ++nt) {
        // B frag contiguous in transposed tile: BsT[n][k], k pair adjacent
        v2f b = *(const v2f*)&BsT[nt * 16 + l16][4 * kk + 2 * half];
        acc[nt] = wmma_f32(a, b, acc[nt]);
      }
    }
    __syncthreads();
  }

#pragma unroll
  for (int nt = 0; nt < 4; ++nt) {
    const int col = colBase + nt * 16 + l16;
    const float bv = bias[col];
#pragma unroll
    for (int j = 0; j < 8; ++j) {   // C layout: VGPRj -> M = j + 8*half
      const int row = rowBase + wave * 16 + j + 8 * half;
      C[(size_t)row * N + col] = acc[nt][j] + bv;
    }
  }
}

// ---------------------------------------------------------------------------
// Attention: one block per (b, window g, head h). 8 waves, 16 q-rows each.
// ---------------------------------------------------------------------------
__global__ __launch_bounds__(256) void attn_kernel(
    const float* __restrict__ Q, const float* __restrict__ KV,
    const float* __restrict__ peq, const float* __restrict__ pekv,
    const float* __restrict__ mask, float* __restrict__ O) {
  __shared__ float ks[K2N][34];        // rotary'd K, padded (read banks 2*l16)
  __shared__ float pst[8][16][18];     // per-wave P transpose staging

  const int idx = blockIdx.x;
  const int h = idx % KNH;
  const int g = (idx / KNH) % KG;
  const int b = idx / (KNH * KG);
  const int gh = g >> 2, gw = g & 3;

  const int tid = threadIdx.x;
  const int wave = tid >> 5;
  const int lane = tid & 31;
  const int l16 = lane & 15;
  const int half = lane >> 4;

  // ---- stage K (roll-gather + rotary) into LDS: one key row per thread ----
  {
    const int j = tid;                 // 0..255
    const int s = j >> 7;
    const int n = j & 127;
    const int hsrc =
        (gh * 8 + (n >> 4) + (s == 0 ? KSHIFT : (KH - KSHIFT))) & (KH - 1);
    const int wsrc = gw * 16 + (n & 15);
    const float* krow =
        KV + (size_t)((b * KH + hsrc) * KW + wsrc) * (2 * KDIM) + h * KHEAD;
    const float* cosr = pekv + ((size_t)(0 * KG + g) * K2N + j) * (KHEAD / 2);
    const float* sinr = pekv + ((size_t)(1 * KG + g) * K2N + j) * (KHEAD / 2);
#pragma unroll
    for (int c = 0; c < 16; ++c) {
      const float x1 = krow[2 * c], x2 = krow[2 * c + 1];
      const float cc = cosr[c], ss = sinr[c];
      ks[j][c]      = x1 * cc - x2 * ss;
      ks[j][c + 16] = x2 * cc + x1 * ss;
    }
  }

  // ---- preload rotary'd Q as WMMA A-fragments (this lane: row M = l16) ----
  const int qn = wave * 16 + l16;      // token within window for this A-row
  const int qhs = gh * 8 + (qn >> 4);
  const int qws = gw * 16 + (qn & 15);
  const float* qrow = Q + (size_t)((b * KH + qhs) * KW + qws) * KDIM + h * KHEAD;
  const float* qcos = peq + ((size_t)(0 * KG + g) * KN + qn) * (KHEAD / 2);
  const float* qsin = peq + ((size_t)(1 * KG + g) * KN + qn) * (KHEAD / 2);
  v2f qf[8];
#pragma unroll
  for (int kk = 0; kk < 8; ++kk) {
    const int k0 = 4 * kk + 2 * half;  // channel of fragment element 0
#pragma unroll
    for (int e = 0; e < 2; ++e) {
      if (kk < 4) {  // out[c] = x1*cos - x2*sin
        const int c = k0 + e;
        qf[kk][e] = qrow[2 * c] * qcos[c] - qrow[2 * c + 1] * qsin[c];
      } else {       // out[c+16] = x2*cos + x1*sin
        const int c = k0 + e - 16;
        qf[kk][e] = qrow[2 * c] * qsin[c] + qrow[2 * c + 1] * qcos[c];
      }
    }
  }

  __syncthreads();  // ks ready

  const float scale = 0.1767766952966369f;  // 1/sqrt(32)
  float mrowv[8], lrowv[8];
  v8f oacc[2];
  oacc[0] = {}; oacc[1] = {};
#pragma unroll
  for (int j = 0; j < 8; ++j) { mrowv[j] = -1e30f; lrowv[j] = 0.f; }

  const float* maskg = mask + (size_t)g * KN * K2N;

  for (int kt = 0; kt < 16; ++kt) {   // 16 key tiles of 16
    v8f s = {};
#pragma unroll
    for (int kk = 0; kk < 8; ++kk) {  // K=32 channels, 4 per WMMA
      // B = K^T: B[c][n] = ks[key=n][c]; contiguous c pair -> single b64
      v2f bfr = *(const v2f*)&ks[kt * 16 + l16][4 * kk + 2 * half];
      s = wmma_f32(qf[kk], bfr, s);
    }

    const int keyc = kt * 16 + l16;   // this lane's key column (C layout N)
#pragma unroll
    for (int j = 0; j < 8; ++j) {
      const int qr = wave * 16 + j + 8 * half;
      s[j] = s[j] * scale + maskg[(size_t)qr * K2N + keyc];
    }

    // online softmax per row (reduction across the 16 lanes of each half)
#pragma unroll
    for (int j = 0; j < 8; ++j) {
      float m = s[j];
#pragma unroll
      for (int d = 1; d < 16; d <<= 1) m = fmaxf(m, __shfl_xor(m, d, 32));
      const float mn = fmaxf(mrowv[j], m);
      const float alpha = expf(mrowv[j] - mn);
      const float pv = expf(s[j] - mn);
      s[j] = pv;
      float rs = pv;
#pragma unroll
      for (int d = 1; d < 16; d <<= 1) rs += __shfl_xor(rs, d, 32);
      lrowv[j] = lrowv[j] * alpha + rs;
      mrowv[j] = mn;
      oacc[0][j] *= alpha;
      oacc[1][j] *= alpha;
    }

    // transpose P (C layout -> A layout) through LDS (same-wave only)
#pragma unroll
    for (int j = 0; j < 8; ++j)
      pst[wave][j + 8 * half][l16] = s[j];
    asm volatile("s_wait_dscnt 0" ::: "memory");

    // O += P(16x16) @ V(16x32); rolled V source row is constant per key-tile
    const int sv   = kt >> 3;
    const int hsrc = (gh * 8 + (kt & 7) + (sv == 0 ? KSHIFT : (KH - KSHIFT))) &
                     (KH - 1);
    const float* vbase = KV +
        (size_t)((b * KH + hsrc) * KW + gw * 16) * (2 * KDIM) + KDIM +
        h * KHEAD;
#pragma unroll
    for (int kk = 0; kk < 4; ++kk) {
      v2f a = *(const v2f*)&pst[wave][l16][4 * kk + 2 * half];
      const float* v0 = vbase + (size_t)(4 * kk + 2 * half) * (2 * KDIM);
#pragma unroll
      for (int ct = 0; ct < 2; ++ct) {
        v2f bv;
        bv.x = v0[ct * 16 + l16];
        bv.y = v0[2 * KDIM + ct * 16 + l16];
        oacc[ct] = wmma_f32(a, bv, oacc[ct]);
      }
    }
  }

  // ---- write O back in (B,H,W,DIM) layout ----
#pragma unroll
  for (int ct = 0; ct < 2; ++ct) {
#pragma unroll
    for (int j = 0; j < 8; ++j) {
      const int n = wave * 16 + j + 8 * half;
      const int hs = gh * 8 + (n >> 4);
      const int ws = gw * 16 + (n & 15);
      O[(size_t)((b * KH + hs) * KW + ws) * KDIM + h * KHEAD + ct * 16 + l16] =
          oacc[ct][j] / lrowv[j];
    }
  }
}

// ---------------------------------------------------------------------------
extern "C" void kernel_launch(void* const* d_in, const int* in_sizes, int n_in,
                              void* d_out, int out_size, void* d_ws,
                              size_t ws_size, hipStream_t stream) {
  const float* x      = (const float*)d_in[0];
  const float* peq    = (const float*)d_in[1];
  const float* pekv   = (const float*)d_in[2];
  const float* mask   = (const float*)d_in[3];
  const float* w_q    = (const float*)d_in[4];
  const float* b_q    = (const float*)d_in[5];
  const float* w_kv   = (const float*)d_in[6];
  const float* b_kv   = (const float*)d_in[7];
  const float* w_proj = (const float*)d_in[8];
  const float* b_proj = (const float*)d_in[9];

  const int B = in_sizes[0] / (KH * KW * KDIM);  // 8
  const int M = B * KH * KW;                     // 32768 tokens

  float* qbuf  = (float*)d_ws;                   //  64 MB
  float* kvbuf = qbuf + (size_t)M * KDIM;        // 128 MB
  float* obuf  = kvbuf + (size_t)M * 2 * KDIM;   //  64 MB

  dim3 blk(256);
  gemm_bias_kernel<<<dim3((M / 128) * (KDIM / 64)), blk, 0, stream>>>(
      x, w_q, b_q, qbuf, M, KDIM);
  gemm_bias_kernel<<<dim3((M / 128) * (2 * KDIM / 64)), blk, 0, stream>>>(
      x, w_kv, b_kv, kvbuf, M, 2 * KDIM);
  attn_kernel<<<dim3(B * KG * KNH), blk, 0, stream>>>(qbuf, kvbuf, peq, pekv,
                                                      mask, obuf);
  gemm_bias_kernel<<<dim3((M / 128) * (KDIM / 64)), blk, 0, stream>>>(
      obuf, w_proj, b_proj, (float*)d_out, M, KDIM);
}